// Signal_Propagation_MLP_ODE_14078902797099
// MI455X (gfx1250) — compile-verified
//
#include <hip/hip_runtime.h>
#include <hip/hip_bf16.h>
#include <math.h>

typedef __attribute__((ext_vector_type(2))) float v2f;
typedef __attribute__((ext_vector_type(8))) float v8f;

#define NN 65536
#define NI 16384
#define HID 512
#define T_END 0.01f
#define RTOLc 1e-5f
#define ATOLc 1e-7f

// ---------------- WMMA f32 GEMV primitives ----------------
// C(16x16) += A(16x4) x B(4x16); only row M=0 of A/C carries the GEMV.
__device__ __forceinline__ v8f wmma_f32(v2f a, v2f b, v8f c) {
  return __builtin_amdgcn_wmma_f32_16x16x4_f32(false, a, false, b, (short)0, c,
                                               false, false);
}

// A fragment: row 0 = v[k..k+3]; lanes 0 and 16 carry data (K split across
// lane halves per the 16x4 F32 A layout). Branchless: every lane loads the
// (half-wave-uniform, cache-hit) pair and selects, so no EXEC save/restore.
__device__ __forceinline__ v2f gemv_afrag(const float* __restrict__ v, int k,
                                          int lane) {
  const float* p = v + k + ((lane >> 4) << 1);
  float a0 = p[0];
  float a1 = p[1];
  bool m = (lane & 15) == 0;
  v2f a;
  a[0] = m ? a0 : 0.f;
  a[1] = m ? a1 : 0.f;
  return a;
}

// ---------------- setup ----------------
__global__ void k_init(const float* __restrict__ x, const float* __restrict__ b1,
                       float* y, float* c1acc, float* sc) {
  int i = blockIdx.x * blockDim.x + threadIdx.x;
  if (i < NN) y[i] = x[(size_t)i * 8 + 3];          // v0 = x[:,3]
  if (i < HID) c1acc[i] = b1[i];
  if (i == 0) { sc[0] = 0.f; sc[1] = T_END; sc[2] = 0.f; sc[3] = 0.f; sc[4] = 0.f; }
}

// Constant input contribution: c1acc[n] += sum_k I[k] * W1[65536+k][n]
__global__ void k_const_in(const float* __restrict__ x,
                           const float* __restrict__ W1, float* c1acc) {
  int n = threadIdx.x;                    // 512 threads
  int span = NI / gridDim.x;
  int k0 = blockIdx.x * span;
  float acc = 0.f;
  for (int k = k0; k < k0 + span; ++k) {
    float Ik = x[(size_t)k * 8 + 4];      // I = x[:n_input,4]
    acc += Ik * W1[(size_t)(NN + k) * HID + n];
  }
  atomicAdd(&c1acc[n], acc);
}

// ---------------- dopri5 scalar control ----------------
__global__ void k_step_begin(float* sc) {
  float t = sc[0], h = sc[1];
  bool done = t >= T_END;
  sc[2] = done ? 0.f : fminf(h, T_END - t);  // h_eff
  sc[3] = 0.f;                               // err accumulator
}

__global__ void k_step_end(float* sc) {
  float t = sc[0], h = sc[1], heff = sc[2];
  float errn = sqrtf(sc[3] / (float)NN + 1e-16f);
  bool done = t >= T_END;
  bool take = (errn <= 1.0f) && !done;
  float fac = 0.9f * powf(errn + 1e-10f, -0.2f);
  fac = fminf(fmaxf(fac, 0.2f), 5.0f);
  sc[0] = take ? t + heff : t;
  sc[1] = done ? h : heff * fac;
  sc[4] = take ? 1.f : 0.f;
}

// dst = y + h_eff*(c1*k1 + ... + c6*k6); also zero layer-1 accumulator.
__global__ void k_stage_prep(const float* __restrict__ y,
                             const float* __restrict__ kb, float* dst,
                             float* h1pre, const float* __restrict__ sc,
                             float c1, float c2, float c3, float c4, float c5,
                             float c6) {
  int i = blockIdx.x * blockDim.x + threadIdx.x;
  float heff = sc[2];
  float acc = 0.f;
  if (c1 != 0.f) acc += c1 * kb[i];
  if (c2 != 0.f) acc += c2 * kb[(size_t)NN + i];
  if (c3 != 0.f) acc += c3 * kb[(size_t)2 * NN + i];
  if (c4 != 0.f) acc += c4 * kb[(size_t)3 * NN + i];
  if (c5 != 0.f) acc += c5 * kb[(size_t)4 * NN + i];
  if (c6 != 0.f) acc += c6 * kb[(size_t)5 * NN + i];
  dst[i] = y[i] + heff * acc;
  if (i < HID) h1pre[i] = 0.f;
}

// ---------------- drift layers (WMMA f32 GEMV) ----------------
// layer1 partial: h1pre[n] += v[0:65536] . W1[0:65536, n]
__global__ void k_layer1(const float* __restrict__ v,
                         const float* __restrict__ W1, float* h1pre) {
  const int lane = threadIdx.x & 31;
  const int wave = __builtin_amdgcn_readfirstlane(threadIdx.x >> 5);
  const int n0 = blockIdx.x * 16;
  const int waves_per_blk = blockDim.x >> 5;
  const int kspan = NN / (gridDim.y * waves_per_blk);   // 1024
  const int k0 = (blockIdx.y * waves_per_blk + wave) * kspan;

  const int ld = HID;
  const float* Wp = W1 + (size_t)(k0 + (lane >> 4) * 2) * ld + n0 + (lane & 15);
  v8f c = {};
  for (int k = k0; k < k0 + kspan; k += 16) {
    __builtin_prefetch(Wp + (size_t)64 * ld, 0, 1);
#pragma unroll
    for (int u = 0; u < 4; ++u) {
      v2f a = gemv_afrag(v, k + 4 * u, lane);
      v2f b;
      b[0] = Wp[0];
      b[1] = Wp[ld];
      c = wmma_f32(a, b, c);
      Wp += (size_t)4 * ld;
    }
  }
  if (lane < 16) atomicAdd(&h1pre[n0 + lane], c[0]);
}

// layer2: h2pre[n] = tanh(h1pre+c1acc) . W2[:,n] + b2[n]
__global__ void k_layer2(const float* __restrict__ h1pre,
                         const float* __restrict__ c1acc,
                         const float* __restrict__ W2,
                         const float* __restrict__ b2, float* h2pre) {
  __shared__ float s_h[HID];
  for (int i = threadIdx.x; i < HID; i += blockDim.x)
    s_h[i] = tanhf(h1pre[i] + c1acc[i]);
  __syncthreads();
  const int lane = threadIdx.x & 31;
  const int wave = __builtin_amdgcn_readfirstlane(threadIdx.x >> 5);
  const int n0 = (blockIdx.x * (blockDim.x >> 5) + wave) * 16;

  const int ld = HID;
  const float* Wp = W2 + (size_t)((lane >> 4) * 2) * ld + n0 + (lane & 15);
  v8f c = {};
  for (int k = 0; k < HID; k += 16) {
#pragma unroll
    for (int u = 0; u < 4; ++u) {
      v2f a = gemv_afrag(s_h, k + 4 * u, lane);
      v2f b;
      b[0] = Wp[0];
      b[1] = Wp[ld];
      c = wmma_f32(a, b, c);
      Wp += (size_t)4 * ld;
    }
  }
  if (lane < 16) h2pre[n0 + lane] = c[0] + b2[n0 + lane];
}

// layer3: kout[n] = tanh(h2pre) . W3[:,n] + b3[n]
__global__ void k_layer3(const float* __restrict__ h2pre,
                         const float* __restrict__ W3,
                         const float* __restrict__ b3, float* kout) {
  __shared__ float s_h[HID];
  for (int i = threadIdx.x; i < HID; i += blockDim.x)
    s_h[i] = tanhf(h2pre[i]);
  __syncthreads();
  const int lane = threadIdx.x & 31;
  const int wave = __builtin_amdgcn_readfirstlane(threadIdx.x >> 5);
  const int n0 = (blockIdx.x * (blockDim.x >> 5) + wave) * 16;

  const int ld = NN;
  const float* Wp = W3 + (size_t)((lane >> 4) * 2) * ld + n0 + (lane & 15);
  v8f c = {};
  for (int k = 0; k < HID; k += 16) {
    __builtin_prefetch(Wp + (size_t)64 * ld, 0, 1);
#pragma unroll
    for (int u = 0; u < 4; ++u) {
      v2f a = gemv_afrag(s_h, k + 4 * u, lane);
      v2f b;
      b[0] = Wp[0];
      b[1] = Wp[ld];
      c = wmma_f32(a, b, c);
      Wp += (size_t)4 * ld;
    }
  }
  if (lane < 16) kout[n0 + lane] = c[0] + b3[n0 + lane];
}

// ---------------- error norm / commit / output ----------------
__global__ void k_err(const float* __restrict__ y, const float* __restrict__ y5,
                      const float* __restrict__ kb, float* sc) {
  const float E1 = 35.0f / 384.0f - 5179.0f / 57600.0f;
  const float E3 = 500.0f / 1113.0f - 7571.0f / 16695.0f;
  const float E4 = 125.0f / 192.0f - 393.0f / 640.0f;
  const float E5 = -2187.0f / 6784.0f + 92097.0f / 339200.0f;
  const float E6 = 11.0f / 84.0f - 187.0f / 2100.0f;
  const float E7 = -1.0f / 40.0f;
  int i = blockIdx.x * blockDim.x + threadIdx.x;
  float heff = sc[2];
  float err = heff * (E1 * kb[i] + E3 * kb[(size_t)2 * NN + i] +
                      E4 * kb[(size_t)3 * NN + i] + E5 * kb[(size_t)4 * NN + i] +
                      E6 * kb[(size_t)5 * NN + i] + E7 * kb[(size_t)6 * NN + i]);
  float scale = ATOLc + RTOLc * fmaxf(fabsf(y[i]), fabsf(y5[i]));
  float r = err / scale;
  float v = r * r;
  for (int o = 16; o > 0; o >>= 1) v += __shfl_down(v, o, 32);
  if ((threadIdx.x & 31) == 0) atomicAdd(&sc[3], v);
}

__global__ void k_commit(float* y, const float* __restrict__ y5,
                         const float* __restrict__ sc) {
  if (sc[4] == 0.f) return;
  int i = blockIdx.x * blockDim.x + threadIdx.x;
  y[i] = y5[i];
}

__global__ void k_final(const float* __restrict__ x, const float* __restrict__ y,
                        float* out) {
  int i = blockIdx.x * blockDim.x + threadIdx.x;
  out[i] = (y[i] - x[(size_t)i * 8 + 3]) / T_END;
}

// ---------------- host orchestration ----------------
extern "C" void kernel_launch(void* const* d_in, const int* in_sizes, int n_in,
                              void* d_out, int out_size, void* d_ws,
                              size_t ws_size, hipStream_t stream) {
  const float* x  = (const float*)d_in[0];
  const float* W1 = (const float*)d_in[1];
  const float* b1 = (const float*)d_in[2];
  const float* W2 = (const float*)d_in[3];
  const float* b2 = (const float*)d_in[4];
  const float* W3 = (const float*)d_in[5];
  const float* b3 = (const float*)d_in[6];
  float* out = (float*)d_out;

  float* w = (float*)d_ws;
  float* y     = w;
  float* y5    = w + (size_t)NN;
  float* ytmp  = w + (size_t)2 * NN;
  float* kbuf  = w + (size_t)3 * NN;       // k1..k7, stride NN
  float* h1pre = w + (size_t)10 * NN;
  float* c1acc = h1pre + HID;
  float* h2pre = c1acc + HID;
  float* sc    = h2pre + HID;              // t, h, h_eff, err_acc, take

  k_init<<<NN / 256, 256, 0, stream>>>(x, b1, y, c1acc, sc);
  k_const_in<<<32, 512, 0, stream>>>(x, W1, c1acc);

  // Butcher tableau: stage-input coefficients on k1..k6 (stage 7 builds y5)
  const float cf[7][6] = {
      {0.f, 0.f, 0.f, 0.f, 0.f, 0.f},
      {1.f / 5.f, 0.f, 0.f, 0.f, 0.f, 0.f},
      {3.f / 40.f, 9.f / 40.f, 0.f, 0.f, 0.f, 0.f},
      {44.f / 45.f, -56.f / 15.f, 32.f / 9.f, 0.f, 0.f, 0.f},
      {19372.f / 6561.f, -25360.f / 2187.f, 64448.f / 6561.f, -212.f / 729.f,
       0.f, 0.f},
      {9017.f / 3168.f, -355.f / 33.f, 46732.f / 5247.f, 49.f / 176.f,
       -5103.f / 18656.f, 0.f},
      {35.f / 384.f, 0.f, 500.f / 1113.f, 125.f / 192.f, -2187.f / 6784.f,
       11.f / 84.f}};

  for (int step = 0; step < 8; ++step) {
    k_step_begin<<<1, 1, 0, stream>>>(sc);
    for (int s = 0; s < 7; ++s) {
      float* dst = (s < 6) ? ytmp : y5;
      k_stage_prep<<<NN / 256, 256, 0, stream>>>(y, kbuf, dst, h1pre, sc,
                                                 cf[s][0], cf[s][1], cf[s][2],
                                                 cf[s][3], cf[s][4], cf[s][5]);
      k_layer1<<<dim3(HID / 16, 8), 256, 0, stream>>>(dst, W1, h1pre);
      k_layer2<<<(HID / 16) / 8, 256, 0, stream>>>(h1pre, c1acc, W2, b2, h2pre);
      k_layer3<<<(NN / 16) / 8, 256, 0, stream>>>(h2pre, W3, b3,
                                                  kbuf + (size_t)s * NN);
    }
    k_err<<<NN / 256, 256, 0, stream>>>(y, y5, kbuf, sc);
    k_step_end<<<1, 1, 0, stream>>>(sc);
    k_commit<<<NN / 256, 256, 0, stream>>>(y, y5, sc);
  }
  k_final<<<NN / 256, 256, 0, stream>>>(x, y, out);
}